// testModule_38397007626958
// MI455X (gfx1250) — compile-verified
//
#include <hip/hip_runtime.h>
#include <hip/hip_bf16.h>

typedef __attribute__((ext_vector_type(2))) float v2f;
typedef __attribute__((ext_vector_type(8))) float v8f;

#define DIM 300   // feature dim D
#define L1N 256
#define L2N 256
#define XSTRIDE 306   // even (8B-aligned B64 rows), 306%64=50 -> distinct banks

// ---------------------------------------------------------------------------
// Kernel 0: linA = input1 @ W_lin[:, :D].T ; linB = input2 @ W_lin[:, D:].T
// ws layout: lin[0..767] = linA[i][k], lin[768..1535] = linB[j][k]
// ---------------------------------------------------------------------------
__global__ __launch_bounds__(256) void lin_kernel(
    const float* __restrict__ input1, const float* __restrict__ input2,
    const float* __restrict__ W_lin, float* __restrict__ lin)
{
    const int row = threadIdx.x;
    const bool isA = (blockIdx.x == 0);
    const float* x = (isA ? input1 : input2) + row * DIM;
    const int off = isA ? 0 : DIM;
    for (int k = 0; k < 3; ++k) {
        float acc = 0.0f;
        const float* w = W_lin + k * (2 * DIM) + off;
        for (int d = 0; d < DIM; ++d) acc += x[d] * w[d];
        lin[blockIdx.x * 768 + row * 3 + k] = acc;
    }
}

// ---------------------------------------------------------------------------
// Kernel 1: S[i,j,k] = relu( (input1 @ T_k @ input2^T)[i,j] + linA[i,k]
//                            + linB[j,k] + b_lin[k] )
// One workgroup per (k, i-tile of 16 rows); 8 waves.
// Stage A: Xtile(16 x 300) = input1[i0:i0+16] @ T_k  -> LDS   (f32 WMMA, K=4)
// Stage B: per j-tile, C = Xtile @ input2[j0:j0+16]^T        (f32 WMMA, K=4)
// Boundary handling is branch-free: clamped index + 0/1 mask multiply, so the
// K-loop is pure global_load_b64 + v_mul + v_wmma (no EXEC juggling).
// ---------------------------------------------------------------------------
__global__ __launch_bounds__(256) void bilinear_wmma_kernel(
    const float* __restrict__ input1, const float* __restrict__ input2,
    const float* __restrict__ T, const float* __restrict__ b_lin,
    const float* __restrict__ lin, float* __restrict__ S)
{
    const int k    = blockIdx.x / 16;
    const int i0   = (blockIdx.x % 16) * 16;
    const int tid  = threadIdx.x;
    const int wave = tid >> 5;
    const int lane = tid & 31;
    const int half = lane >> 4;      // 0: lanes 0-15, 1: lanes 16-31
    const int l    = lane & 15;

    __shared__ float Xs[16 * XSTRIDE];

    const float* Tk = T + k * (DIM * DIM);
    const float* Arow = input1 + (i0 + l) * DIM;   // A-fragment source row

    // ---------------- Stage A: Xtile = input1_tile @ T_k ----------------
    // 19 column tiles of 16 (cols 288..303 masked to zero past DIM=300)
    for (int ct = wave; ct < 19; ct += 8) {
        v8f acc = {};
        const int e0   = ct * 16 + l;                  // output column (N)
        const int ec   = (e0 < DIM) ? e0 : (DIM - 1);  // clamped (always legal)
        const float mk = (e0 < DIM) ? 1.0f : 0.0f;     // hoisted 0/1 mask
        const float* Bcol = Tk + ec;
        for (int kk = 0; kk < DIM; kk += 4) {
            const int d0 = kk + (half ? 2 : 0);        // VGPR0 K-index
            v2f a, b;
            // A 16x4: lanes 0-15 & 16-31 hold M=0..15; v0:K={0|2}, v1:K={1|3}
            a.x = Arow[d0];
            a.y = Arow[d0 + 1];
            // B 4x16: v0 rows K={0|2}, v1 rows K={1|3}; lane%16 = N
            b.x = Bcol[(d0    ) * DIM] * mk;
            b.y = Bcol[(d0 + 1) * DIM] * mk;
            acc = __builtin_amdgcn_wmma_f32_16x16x4_f32(
                false, a, false, b, (short)0, acc, false, false);
        }
        // C 16x16: vgpr r, lanes 0-15 -> M=r, lanes 16-31 -> M=r+8, N=lane%16
        for (int r = 0; r < 8; ++r)
            Xs[(r + 8 * half) * XSTRIDE + ct * 16 + l] = acc[r];
    }
    __syncthreads();

    // ---------------- Stage B: bil tile + fused epilogue ----------------
    const float blk = b_lin[k];
    for (int jt = wave; jt < 16; jt += 8) {
        v8f acc = {};
        const int j0 = jt * 16;
        const float* Brow = input2 + (j0 + l) * DIM;   // B[e,j] = input2[j,e]
        for (int kk = 0; kk < DIM; kk += 4) {
            const int e0 = kk + (half ? 2 : 0);
            v2f a, b;
            a.x = Xs[l * XSTRIDE + e0];        // 8B-aligned ds_load_b64
            a.y = Xs[l * XSTRIDE + e0 + 1];
            b.x = Brow[e0];
            b.y = Brow[e0 + 1];
            acc = __builtin_amdgcn_wmma_f32_16x16x4_f32(
                false, a, false, b, (short)0, acc, false, false);
        }
        for (int r = 0; r < 8; ++r) {
            const int i = i0 + r + 8 * half;
            const int j = j0 + l;
            float v = acc[r] + lin[i * 3 + k] + lin[768 + j * 3 + k] + blk;
            v = fmaxf(v, 0.0f);
            S[(i * L2N + j) * 3 + k] = v;
        }
    }
}

// ---------------------------------------------------------------------------
// Kernel 2: 2D-grid GRU recurrence, anti-diagonal wavefront.
// One workgroup, one thread per column j. Hu[j] = h from last diagonal,
// Hd[j] = h from the diagonal before that. 511 steps, 2 barriers each.
// ---------------------------------------------------------------------------
__global__ __launch_bounds__(256) void grid_gru_kernel(
    const float* __restrict__ S,
    const float* __restrict__ Wqr, const float* __restrict__ bqr,
    const float* __restrict__ Wqz, const float* __restrict__ bqz,
    const float* __restrict__ Uw,  const float* __restrict__ Wh,
    const float* __restrict__ bh,  const float* __restrict__ Wlast,
    const float* __restrict__ blast, float* __restrict__ out)
{
    __shared__ float Hu[L2N * 3];
    __shared__ float Hd[L2N * 3];
    __shared__ float sWqr[9 * 12], sbqr[9];
    __shared__ float sWqz[12 * 12], sbqz[12];
    __shared__ float sU[3 * 9], sWh[3 * 3], sbh[3];

    const int tid = threadIdx.x;
    if (tid < 108) sWqr[tid] = Wqr[tid];
    if (tid < 144) sWqz[tid] = Wqz[tid];
    if (tid < 27)  sU[tid]   = Uw[tid];
    if (tid < 12)  sbqz[tid] = bqz[tid];
    if (tid < 9) { sbqr[tid] = bqr[tid]; sWh[tid] = Wh[tid]; }
    if (tid < 3)   sbh[tid]  = bh[tid];
    for (int c = 0; c < 3; ++c) { Hu[tid * 3 + c] = 0.0f; Hd[tid * 3 + c] = 0.0f; }
    __syncthreads();

    const int j = tid;
    for (int d = 0; d < 511; ++d) {
        const int i = d - j;
        const bool active = (i >= 0) && (i < L1N);
        float hup[3], hleft[3], hdiag[3], hnew[3];
        if (active) {
            for (int c = 0; c < 3; ++c) {
                hup[c]   = Hu[j * 3 + c];
                hleft[c] = (j > 0)          ? Hu[(j - 1) * 3 + c] : 0.0f;
                hdiag[c] = (j > 0 && i > 0) ? Hd[(j - 1) * 3 + c] : 0.0f;
            }
            float q[12];
            for (int c = 0; c < 3; ++c) {
                q[c]     = hup[c];
                q[3 + c] = hleft[c];
                q[6 + c] = hdiag[c];
            }
            const float* sp = S + (i * L2N + j) * 3;
            q[9] = sp[0]; q[10] = sp[1]; q[11] = sp[2];

            float r[9];
            for (int t = 0; t < 9; ++t) {
                float a = sbqr[t];
                for (int c = 0; c < 12; ++c) a += sWqr[t * 12 + c] * q[c];
                r[t] = 1.0f / (1.0f + expf(-a));
            }
            float z[12];
            for (int t = 0; t < 12; ++t) {
                float a = sbqz[t];
                for (int c = 0; c < 12; ++c) a += sWqz[t * 12 + c] * q[c];
                z[t] = a;
            }
            // gates: per hidden unit t, softmax over {z[t], z[3+t], z[6+t], z[9+t]}
            float g[4][3];
            for (int t = 0; t < 3; ++t) {
                float m = z[t];
                for (int gg = 1; gg < 4; ++gg) m = fmaxf(m, z[gg * 3 + t]);
                float e[4], ssum = 0.0f;
                for (int gg = 0; gg < 4; ++gg) { e[gg] = expf(z[gg * 3 + t] - m); ssum += e[gg]; }
                for (int gg = 0; gg < 4; ++gg) g[gg][t] = e[gg] / ssum;
            }
            float rh[9];
            for (int t = 0; t < 9; ++t) rh[t] = r[t] * q[t];
            for (int t = 0; t < 3; ++t) {
                float a = sbh[t];
                for (int c = 0; c < 3; ++c) a += sWh[t * 3 + c] * q[9 + c];
                for (int c = 0; c < 9; ++c) a += sU[t * 9 + c] * rh[c];
                float ht = tanhf(a);
                hnew[t] = g[1][t] * hleft[t] + g[2][t] * hup[t]
                        + g[3][t] * hdiag[t] + g[0][t] * ht;
            }
        }
        __syncthreads();
        if (active) {
            for (int c = 0; c < 3; ++c) {
                Hd[j * 3 + c] = hup[c];   // value from two diagonals back
                Hu[j * 3 + c] = hnew[c];
            }
        }
        __syncthreads();
    }

    if (tid == 0) {
        float h0 = Hu[255 * 3 + 0], h1 = Hu[255 * 3 + 1], h2 = Hu[255 * 3 + 2];
        float lg[2];
        for (int t = 0; t < 2; ++t)
            lg[t] = blast[t] + Wlast[t * 3 + 0] * h0 + Wlast[t * 3 + 1] * h1
                             + Wlast[t * 3 + 2] * h2;
        float m = fmaxf(lg[0], lg[1]);
        float e0 = expf(lg[0] - m), e1 = expf(lg[1] - m);
        out[0] = e0 / (e0 + e1);
        out[1] = e1 / (e0 + e1);
    }
}

// ---------------------------------------------------------------------------
extern "C" void kernel_launch(void* const* d_in, const int* in_sizes, int n_in,
                              void* d_out, int out_size, void* d_ws, size_t ws_size,
                              hipStream_t stream) {
    const float* input1 = (const float*)d_in[0];
    const float* input2 = (const float*)d_in[1];
    const float* T      = (const float*)d_in[2];
    const float* W_lin  = (const float*)d_in[3];
    const float* b_lin  = (const float*)d_in[4];
    const float* Wqr    = (const float*)d_in[5];
    const float* bqr    = (const float*)d_in[6];
    const float* Wqz    = (const float*)d_in[7];
    const float* bqz    = (const float*)d_in[8];
    const float* U      = (const float*)d_in[9];
    const float* Wh     = (const float*)d_in[10];
    const float* bh     = (const float*)d_in[11];
    const float* Wlast  = (const float*)d_in[12];
    const float* blast  = (const float*)d_in[13];

    float* lin = (float*)d_ws;        // 1536 floats (linA | linB)
    float* S   = lin + 1536;          // 256*256*3 floats

    lin_kernel<<<dim3(2), dim3(256), 0, stream>>>(input1, input2, W_lin, lin);
    bilinear_wmma_kernel<<<dim3(48), dim3(256), 0, stream>>>(
        input1, input2, T, b_lin, lin, S);
    grid_gru_kernel<<<dim3(1), dim3(256), 0, stream>>>(
        S, Wqr, bqr, Wqz, bqz, U, Wh, bh, Wlast, blast, (float*)d_out);
}